// RelativeBucketedTimeAndPositionBias_30090540876152
// MI455X (gfx1250) — compile-verified
//
#include <hip/hip_runtime.h>
#include <hip/hip_bf16.h>
#include <stdint.h>

// RelativeBucketedTimeAndPositionBias for MI455X (gfx1250)
// out[b,i,j] = pos_w[N-1 + j - i] + ts_w[clip(trunc(log(clip(|ext[b,i+1]-ts[b,j]|,1))/0.301),0,128)]
// Store-bandwidth-bound (256MB out > 192MB L2): NT store hints + async-to-LDS staging.

#define N        4096
#define ROWS     16          // rows per block
#define TPB      256         // 8 wave32s
#define NBUCKMAX 129
// ln(2) / 0.301  (hardware v_log_f32 is log2; fold base conversion into one constant)
#define L2SCALE  2.3028145f

typedef float v4f __attribute__((ext_vector_type(4)));

// gfx1250 async DMA: copy 16 bytes global -> LDS, tracked by ASYNCcnt.
// LDS address operand is the low 32 bits of the generic pointer (ISA 10.2 aperture rule).
__device__ __forceinline__ void async_cp16(const void* g, void* l) {
    asm volatile("global_load_async_to_lds_b128 %0, %1, off"
                 :: "v"((uint32_t)(uintptr_t)l), "v"(g)
                 : "memory");
}

__device__ __forceinline__ float bucket_w(int d, const float* tsw, int nb) {
    int m = d < 0 ? -d : d;
    m = m > 1 ? m : 1;                           // clip(|diff|, 1, ...)
    int bk = (int)(__log2f((float)m) * L2SCALE); // trunc; log2>=0 so trunc==floor
    bk = bk < 0 ? 0 : (bk > nb ? nb : bk);
    return tsw[bk];
}

__global__ __launch_bounds__(TPB)
void bias_kernel(const int* __restrict__ ts,
                 const float* __restrict__ ts_w,
                 const float* __restrict__ pos_w,
                 float* __restrict__ out,
                 int nbuckets)
{
    __shared__ int   s_ts[N];            // 16 KB: this batch's timestamp row
    __shared__ float s_posw[2 * N];      // 32 KB: full toeplitz weights (8191 + 1 pad)
    __shared__ float s_tsw[NBUCKMAX];    // bucket weights

    const int tid  = threadIdx.x;
    const int nrb  = N / ROWS;
    const int b    = blockIdx.x / nrb;
    const int row0 = (blockIdx.x % nrb) * ROWS;
    const int* tsrow = ts + b * N;

    // ---- stage inputs into LDS with async-to-LDS DMA (128b chunks, unrolled) ----
#pragma unroll
    for (int u = 0; u < (N / 4) / TPB; ++u) {           // 4 chunks/thread, no guard
        const int c = u * TPB + tid;
        async_cp16(tsrow + 4 * c, s_ts + 4 * c);
    }
#pragma unroll
    for (int u = 0; u < 8; ++u) {                        // 2047 chunks -> floats [0,8188)
        const int c = u * TPB + tid;
        if (c < (2 * N) / 4 - 1)
            async_cp16(pos_w + 4 * c, s_posw + 4 * c);
    }
    if (tid < 3)  s_posw[2 * N - 4 + tid] = pos_w[2 * N - 4 + tid]; // 8188..8190
    if (tid == 3) s_posw[2 * N - 1] = 0.0f;                          // pad (never read)
    if (tid <= nbuckets && tid < NBUCKMAX) s_tsw[tid] = ts_w[tid];

    asm volatile("s_wait_asynccnt 0" ::: "memory");
    __syncthreads();

    // a[r] = ext[b, i+1]  (ext[N] duplicates ts[N-1])
    int aval[ROWS];
#pragma unroll
    for (int r = 0; r < ROWS; ++r) {
        int i   = row0 + r;
        int ip1 = (i + 1 < N) ? (i + 1) : (N - 1);
        aval[r] = s_ts[ip1];
    }

    // Each k-iteration: one 128b LDS load of 4 ts values, reused across all 16 rows.
    for (int k = 0; k < N / (TPB * 4); ++k) {            // 4 iterations
        const int  j  = (k * TPB + tid) * 4;
        const int4 t4 = *reinterpret_cast<const int4*>(&s_ts[j]);
#pragma unroll
        for (int r = 0; r < ROWS; ++r) {
            const int i    = row0 + r;
            const int a    = aval[r];
            const int base = (N - 1) - i + j;            // toeplitz index, contiguous in j
            v4f o;
            o.x = bucket_w(a - t4.x, s_tsw, nbuckets) + s_posw[base + 0];
            o.y = bucket_w(a - t4.y, s_tsw, nbuckets) + s_posw[base + 1];
            o.z = bucket_w(a - t4.z, s_tsw, nbuckets) + s_posw[base + 2];
            o.w = bucket_w(a - t4.w, s_tsw, nbuckets) + s_posw[base + 3];
            const size_t off = ((size_t)(b * N + i) << 12) + (size_t)j;  // *N, N=4096
            // 256MB write-once stream > 192MB L2: non-temporal 128b stores.
            __builtin_nontemporal_store(o, reinterpret_cast<v4f*>(out + off));
        }
    }
}

extern "C" void kernel_launch(void* const* d_in, const int* in_sizes, int n_in,
                              void* d_out, int out_size, void* d_ws, size_t ws_size,
                              hipStream_t stream) {
    const int*   ts    = (const int*)d_in[0];    // [B, N] timestamps (fit in int32)
    const float* ts_w  = (const float*)d_in[1];  // [num_buckets+1]
    const float* pos_w = (const float*)d_in[2];  // [2N-1]
    float*       out   = (float*)d_out;          // [B, N, N] f32

    const int B        = in_sizes[0] / N;
    const int nbuckets = in_sizes[1] - 1;

    dim3 grid(B * (N / ROWS));
    bias_kernel<<<grid, TPB, 0, stream>>>(ts, ts_w, pos_w, out, nbuckets);
}